// DiffMC_27960237097002
// MI455X (gfx1250) — compile-verified
//
#include <hip/hip_runtime.h>
#include <hip/hip_bf16.h>
#include <stdint.h>

// ---------------- problem constants (DIM=64, ISO=0) ----------------
#define NVX   65                       // voxels per axis (DIM+1)
#define MTET  (6 * NVX * NVX * NVX)    // 1,647,750 tets
#define VERTS_F ((size_t)MTET * 18)    // 29,659,500 floats
#define TRIS_F  ((size_t)MTET * 6)     //  9,886,500 floats
#define MASK_F  ((size_t)MTET * 2)     //  3,295,500 floats

#define AS1 __attribute__((address_space(1)))
#define AS3 __attribute__((address_space(3)))

// Vector element types matching the async-LDS builtin signatures
// ('__attribute__((__vector_size__(N * sizeof(int)))) int').
typedef int vli2 __attribute__((vector_size(2 * sizeof(int))));
typedef int vli4 __attribute__((vector_size(4 * sizeof(int))));

#if __has_builtin(__builtin_amdgcn_global_load_async_to_lds_b64) && \
    __has_builtin(__builtin_amdgcn_global_load_async_to_lds_b128) && \
    __has_builtin(__builtin_amdgcn_s_wait_asynccnt)
#define USE_ASYNC_LD 1
#else
#define USE_ASYNC_LD 0
#endif

#if __has_builtin(__builtin_amdgcn_global_store_async_from_lds_b128) && \
    __has_builtin(__builtin_amdgcn_s_wait_asynccnt)
#define USE_ASYNC_ST 1
#else
#define USE_ASYNC_ST 0
#endif

// ---------------- marching-tetrahedra tables ----------------
__device__ __constant__ int c_off[8][3] = {
  {0,0,0},{1,0,0},{1,1,0},{0,1,0},{0,0,1},{1,0,1},{1,1,1},{0,1,1}};
__device__ __constant__ int c_tets[6][4] = {
  {0,1,2,6},{0,2,3,6},{0,3,7,6},{0,7,4,6},{0,4,5,6},{0,5,1,6}};
__device__ __constant__ int c_ea[6] = {0,0,0,1,1,2};
__device__ __constant__ int c_eb[6] = {1,2,3,2,3,3};
__device__ __constant__ int c_tri[16][6] = {
  {-1,-1,-1,-1,-1,-1},{ 0, 1, 2,-1,-1,-1},{ 0, 4, 3,-1,-1,-1},{ 1, 2, 4, 1, 4, 3},
  { 1, 3, 5,-1,-1,-1},{ 0, 3, 5, 0, 5, 2},{ 0, 4, 5, 0, 5, 1},{ 2, 4, 5,-1,-1,-1},
  { 2, 5, 4,-1,-1,-1},{ 0, 5, 4, 0, 1, 5},{ 0, 5, 3, 0, 2, 5},{ 1, 5, 3,-1,-1,-1},
  { 1, 4, 2, 1, 3, 4},{ 0, 3, 4,-1,-1,-1},{ 0, 2, 1,-1,-1,-1},{-1,-1,-1,-1,-1,-1}};

// tris section is a pure iota (arange as int32 -> float)
__global__ void diffmc_tris_iota(float* __restrict__ t, int nq) {
  int i = blockIdx.x * blockDim.x + threadIdx.x;
  if (i < nq) {
    float b = (float)(4 * i);
    float4 v = make_float4(b, b + 1.0f, b + 2.0f, b + 3.0f);
    reinterpret_cast<float4*>(t)[i] = v;
  }
}

// One block per (x,y) voxel row: 65 voxels -> 390 tets. 416 threads = 13 wave32s.
__global__ __launch_bounds__(416) void diffmc_mc_kernel(
    const float* __restrict__ grid,     // [64,64,64]
    const float* __restrict__ deform,   // [64,64,64,3]
    float* __restrict__ out) {
  // LDS layouts chosen so z=1 lands on 8B/16B boundaries for async lane writes.
  // Gs row: [0]=pad z=0, [2+ (z-1)] for z=1..65  (index 66 max), stride 68 floats.
  // Ds row: [0..2]=pad z=0, [4 + (z-1)*3 + c] for z=1..65 (max 198), stride 200 floats.
  __shared__ __align__(16) float Gs[4][68];
  __shared__ __align__(16) float Ds[4][200];
  __shared__ __align__(16) float Vs[390 * 18];   // 28,080 B staged verts

  const int tid = threadIdx.x;
  const int bx  = blockIdx.x;            // bx = x*65 + y
  const int x   = bx / NVX;
  const int y   = bx - x * NVX;

  // ---- phase 0: fill pads (scalar pad = iso+1 = 1.0, deform pad = 0) ----
  for (int i = tid; i < 4 * 68;  i += 416) (&Gs[0][0])[i] = 1.0f;
  for (int i = tid; i < 4 * 200; i += 416) (&Ds[0][0])[i] = 0.0f;
  __syncthreads();

  // ---- phase 1: stage interior rows via async LDS DMA ----
  if (tid < 128) {                        // grid rows: 4 rows x 32 lanes x b64
    int r = tid >> 5, l = tid & 31;
    int gx = x + (r >> 1), gy = y + (r & 1);
    if (gx >= 1 && gx <= 64 && gy >= 1 && gy <= 64) {
      const float* src = grid + ((size_t)(gx - 1) * 64 + (gy - 1)) * 64 + 2 * l;
      float* dst = &Gs[r][2 + 2 * l];
#if USE_ASYNC_LD
      __builtin_amdgcn_global_load_async_to_lds_b64(
          (AS1 vli2*)src, (AS3 vli2*)dst, 0, 0);
#else
      dst[0] = src[0]; dst[1] = src[1];
#endif
    }
  } else if (tid < 320) {                 // deform rows: 4 rows x 48 lanes x b128
    int q = tid - 128;
    int r = q / 48, l = q - r * 48;
    int gx = x + (r >> 1), gy = y + (r & 1);
    if (gx >= 1 && gx <= 64 && gy >= 1 && gy <= 64) {
      const float* src = deform + ((size_t)(gx - 1) * 64 + (gy - 1)) * 192 + 4 * l;
      float* dst = &Ds[r][4 + 4 * l];
#if USE_ASYNC_LD
      __builtin_amdgcn_global_load_async_to_lds_b128(
          (AS1 vli4*)src, (AS3 vli4*)dst, 0, 0);
#else
      reinterpret_cast<float4*>(dst)[0] = reinterpret_cast<const float4*>(src)[0];
#endif
    }
  }
#if USE_ASYNC_LD
  __builtin_amdgcn_s_wait_asynccnt(0);
#endif
  __syncthreads();

  // ---- phase 2: one thread per tet ----
  if (tid < 390) {
    const int vz = tid / 6;
    const int tt = tid - vz * 6;

    float tv[4], px[4], py[4], pz[4];
#pragma unroll
    for (int j = 0; j < 4; ++j) {
      int c  = c_tets[tt][j];
      int dx = c_off[c][0], dy = c_off[c][1], dz = c_off[c][2];
      int r  = dx * 2 + dy;
      int z  = vz + dz;                       // padded-grid z coordinate, 0..65
      int gi = (z == 0) ? 0 : (2 + (z - 1));
      int di = (z == 0) ? 0 : (4 + (z - 1) * 3);
      tv[j] = Gs[r][gi];
      px[j] = (float)(x + dx) + Ds[r][di + 0];
      py[j] = (float)(y + dy) + Ds[r][di + 1];
      pz[j] = (float)z        + Ds[r][di + 2];
    }

    int code = (tv[0] > 0.0f ? 1 : 0) | (tv[1] > 0.0f ? 2 : 0) |
               (tv[2] > 0.0f ? 4 : 0) | (tv[3] > 0.0f ? 8 : 0);

    float* vout = &Vs[tid * 18];
    const float inv_scale = 1.0f / 63.0f;
#pragma unroll
    for (int s = 0; s < 6; ++s) {
      int e = c_tri[code][s];
      float ox = 0.0f, oy = 0.0f, oz = 0.0f;
      if (e >= 0) {
        int a = c_ea[e], b = c_eb[e];
        float va = tv[a], vb = tv[b];
        float den = vb - va;
        float dsafe = (fabsf(den) > 1e-8f) ? den : 1.0f;
        float t = (0.0f - va) / dsafe;
        t = fminf(fmaxf(t, 0.0f), 1.0f);
        ox = (px[a] + t * (px[b] - px[a]) - 1.0f) * inv_scale;
        oy = (py[a] + t * (py[b] - py[a]) - 1.0f) * inv_scale;
        oz = (pz[a] + t * (pz[b] - pz[a]) - 1.0f) * inv_scale;
      }
      vout[s * 3 + 0] = ox;
      vout[s * 3 + 1] = oy;
      vout[s * 3 + 2] = oz;
    }

    // mask: slot validity of tri entries 0 and 3 (2 triangles per tet)
    int m = bx * 390 + tid;                 // global tet id
    float2 mv = make_float2(c_tri[code][0] >= 0 ? 1.0f : 0.0f,
                            c_tri[code][3] >= 0 ? 1.0f : 0.0f);
    reinterpret_cast<float2*>(out + VERTS_F + TRIS_F + (size_t)2 * m)[0] = mv;
  }
  __syncthreads();

  // ---- phase 3: push contiguous 28,080B vert span via async LDS->global ----
  float* og = out + (size_t)bx * 7020;      // 390 tets * 18 floats, 16B aligned
  for (int i = tid; i < 1755; i += 416) {   // 1755 x 16B = 28,080B
#if USE_ASYNC_ST
    __builtin_amdgcn_global_store_async_from_lds_b128(
        (AS1 vli4*)(og + 4 * i), (AS3 vli4*)(Vs + 4 * i), 0, 0);
#else
    reinterpret_cast<float4*>(og)[i] = reinterpret_cast<const float4*>(Vs)[i];
#endif
  }
  // S_ENDPGM performs an implicit wait-idle, draining ASYNCcnt.
}

extern "C" void kernel_launch(void* const* d_in, const int* in_sizes, int n_in,
                              void* d_out, int out_size, void* d_ws, size_t ws_size,
                              hipStream_t stream) {
  const float* grid   = (const float*)d_in[0];
  const float* deform = (const float*)d_in[1];
  float* out = (float*)d_out;

  // tris iota section (9,886,500 floats = 2,471,625 float4s)
  const int nq = (int)(TRIS_F / 4);
  diffmc_tris_iota<<<(nq + 255) / 256, 256, 0, stream>>>(out + VERTS_F, nq);

  // main marching-tets kernel: one block per (x,y) voxel row
  diffmc_mc_kernel<<<NVX * NVX, 416, 0, stream>>>(grid, deform, out);
}